// GaussianMomentsEmbedding_57750130262465
// MI455X (gfx1250) — compile-verified
//
#include <hip/hip_runtime.h>

typedef __attribute__((ext_vector_type(2))) float v2f;
typedef __attribute__((ext_vector_type(8))) float v8f;

#define AFVS 16
#define RADIALB 8
#define NCH 7
#define NY 16
#define RHO_PER_ATOM 112   // NCH * NY
#define OUT_PER_ATOM 443   // 16 + 7 + 3*28 + 4*84
#define CBLK 128
#define EDGE_BLOCKS 2048

// ---------------- spherical harmonics (matches reference _sph_components) ----
__device__ __forceinline__ void sph16(float x, float y, float z, float* Y) {
  Y[0] = 0.28209479177387814f;
  const float c1 = 0.4886025119029199f;
  Y[1] = c1 * y; Y[2] = c1 * z; Y[3] = c1 * x;
  const float c2a = 1.0925484305920792f;
  const float c2b = 0.31539156525252005f;
  const float c2c = 0.5462742152960396f;
  Y[4] = c2a * x * y;
  Y[5] = c2a * y * z;
  Y[6] = c2b * (3.0f * z * z - 1.0f);
  Y[7] = c2a * x * z;
  Y[8] = c2c * (x * x - y * y);
  Y[9]  = 0.5900435899266435f * y * (3.0f * x * x - y * y);
  Y[10] = 2.890611442640554f * x * y * z;
  Y[11] = 0.4570457994644658f * y * (5.0f * z * z - 1.0f);
  Y[12] = 0.3731763325901154f * z * (5.0f * z * z - 3.0f);
  Y[13] = 0.4570457994644658f * x * (5.0f * z * z - 1.0f);
  Y[14] = 1.445305721320277f * z * (x * x - y * y);
  Y[15] = 0.5900435899266435f * x * (x * x - 3.0f * y * y);
}

// ---------------- workspace zeroing ----------------------------------------
__global__ void zero_f32(float* __restrict__ p, long n) {
  long i = (long)blockIdx.x * blockDim.x + threadIdx.x;
  long stride = (long)gridDim.x * blockDim.x;
  for (; i < n; i += stride) p[i] = 0.0f;
}

// ---------------- Gaunt tensors via 16-pt Gauss-Legendre x 32-phi quadrature
__constant__ float GLX[8] = {
  0.09501250983763744f, 0.28160355077925891f, 0.45801677765722739f,
  0.61787624440264375f, 0.75540440835500303f, 0.86563120238783174f,
  0.94457502307323258f, 0.98940093499164993f };
__constant__ float GLW[8] = {
  0.18945061045506850f, 0.18260341504492359f, 0.16915651939500254f,
  0.14959598881657673f, 0.12462897125553387f, 0.09515851168249278f,
  0.06225352393864789f, 0.02715245941175409f };

// Wg layout: [0,45) W112 (3,3,5) | [45,170) W222 (5,5,5)
//            [170,275) W123 (3,5,7) | [275,520) W233 (5,7,7)
__global__ void gaunt_init(float* __restrict__ Wg) {
  __shared__ float sY[512][16];
  __shared__ float sW[512];
  const float TWO_PI = 6.2831853071795864769f;
  for (int t = threadIdx.x; t < 512; t += blockDim.x) {
    int iu = t >> 5, ip = t & 31;
    float u = GLX[iu & 7] * ((iu < 8) ? 1.0f : -1.0f);
    float w = GLW[iu & 7];
    float st = sqrtf(fmaxf(1.0f - u * u, 0.0f));
    float phi = TWO_PI * (float)ip / 32.0f;
    float x = st * __cosf(phi), y = st * __sinf(phi), z = u;
    sph16(x, y, z, &sY[t][0]);
    sW[t] = w * (TWO_PI / 32.0f);
  }
  __syncthreads();
  for (int idx = threadIdx.x; idx < 520; idx += blockDim.x) {
    int a, b, c, oa, ob, oc;
    if (idx < 45)       { int l = idx;       a = l / 15; b = (l % 15) / 5; c = l % 5; oa = 1; ob = 1; oc = 4; }
    else if (idx < 170) { int l = idx - 45;  a = l / 25; b = (l % 25) / 5; c = l % 5; oa = 4; ob = 4; oc = 4; }
    else if (idx < 275) { int l = idx - 170; a = l / 35; b = (l % 35) / 7; c = l % 7; oa = 1; ob = 4; oc = 9; }
    else                { int l = idx - 275; a = l / 49; b = (l % 49) / 7; c = l % 7; oa = 4; ob = 9; oc = 9; }
    float acc = 0.0f;
    for (int p = 0; p < 512; ++p)
      acc += sY[p][oa + a] * sY[p][ob + b] * sY[p][oc + c] * sW[p];
    Wg[idx] = acc;
  }
}

// ---------------- edge kernel: persistent WMMA f32 GEMM + balanced scatter --
// per wave-tile: 16 edges (M), K=128 (=16 species x 8 radial), N=16 (7 real)
// grid-stride over tiles; each wave owns its LDS slice (LDS in-order per wave,
// DScnt) so no block barriers are needed inside the tile loop.
__global__ void edge_wmma(const float* __restrict__ species,
                          const float* __restrict__ radial,
                          const float* __restrict__ swit,
                          const float* __restrict__ vec,
                          const float* __restrict__ chemrad,
                          const int* __restrict__ esrc,
                          const int* __restrict__ edst,
                          float* __restrict__ rho,
                          int nE, int nTiles) {
  __shared__ float sWp[128][16];        // padded weight matrix [K=128][N=16]
  __shared__ float sSpec[4][16][16];    // per-wave: [edge][afv]
  __shared__ float sRad[4][16][8];      // per-wave: [edge][radial]
  __shared__ float sYe[4][16][16];      // per-wave: [edge][Y comp]
  __shared__ float sXs[4][16][8];       // per-wave: xij*switch [edge][chan] (pad 8)
  __shared__ float sSw[4][16];
  __shared__ int   sSrc[4][16];
  __shared__ int   sDst[4][16];

  const int tid  = threadIdx.x;
  const int wave = tid >> 5;
  const int lane = tid & 31;
  const int half = lane >> 4;
  const int m    = lane & 15;

  // weights once per block: chemrad [16][8][7] -> W[k=i*8+j][n], zero-pad n>=7
  for (int t = tid; t < 128 * 16; t += blockDim.x) {
    int kidx = t >> 4, n = t & 15;
    sWp[kidx][n] = (n < NCH) ? chemrad[kidx * NCH + n] : 0.0f;
  }
  __syncthreads();

  for (int tile = blockIdx.x * 4 + wave; tile < nTiles; tile += EDGE_BLOCKS * 4) {
    const int base = tile * 16;

    // ---- stage per-edge scalars + harmonics (lanes 0..15, one edge each)
    if (lane < 16) {
      int e = base + lane; if (e >= nE) e = nE - 1;
      sSrc[wave][lane] = esrc[e];
      sDst[wave][lane] = edst[e];
      sSw[wave][lane]  = swit[e];
      float vx = vec[3 * e + 0], vy = vec[3 * e + 1], vz = vec[3 * e + 2];
      float dinv = rsqrtf(vx * vx + vy * vy + vz * vz);
      float Y[16];
      sph16(vx * dinv, vy * dinv, vz * dinv, Y);
#pragma unroll
      for (int q = 0; q < 16; ++q) sYe[wave][lane][q] = Y[q];
    }
    // radial (16 edges x 8 comps)
    for (int t = lane; t < 128; t += 32) {
      int el = t >> 3, j = t & 7;
      int e = base + el; if (e >= nE) e = nE - 1;
      sRad[wave][el][j] = radial[(long)e * RADIALB + j];
    }
    __builtin_amdgcn_wave_barrier();   // sDst visible to whole wave
    // species gather (16 edges x 16 comps) via staged dst indices
    for (int t = lane; t < 256; t += 32) {
      int el = t >> 4, c = t & 15;
      sSpec[wave][el][c] = species[(long)sDst[wave][el] * AFVS + c];
    }
    __builtin_amdgcn_wave_barrier();

    // ---- WMMA f32 16x16x4, K-loop over 128
    // ISA 7.12.2: A 16x4 f32: lanes[0:16) K={k0,k0+1}, lanes[16:32) K={k0+2,k0+3}
    v8f acc = {0.f, 0.f, 0.f, 0.f, 0.f, 0.f, 0.f, 0.f};
#pragma unroll 4
    for (int k0 = 0; k0 < 128; k0 += 4) {
      int ka = k0 + 2 * half;
      v2f A, B;
      A.x = sSpec[wave][m][ka >> 3]       * sRad[wave][m][ka & 7];
      A.y = sSpec[wave][m][(ka + 1) >> 3] * sRad[wave][m][(ka + 1) & 7];
      B.x = sWp[ka][m];
      B.y = sWp[ka + 1][m];
      acc = __builtin_amdgcn_wmma_f32_16x16x4_f32(
          /*neg_a=*/false, A, /*neg_b=*/false, B,
          /*c_mod=*/(short)0, acc, /*reuse_a=*/false, /*reuse_b=*/false);
    }
    __builtin_amdgcn_wave_barrier();

    // ---- stage xij*switch: lane channel m (<7), edges r+8*half
    if (m < NCH) {
#pragma unroll
      for (int r = 0; r < 8; ++r) {
        int el = r + 8 * half;
        sXs[wave][el][m] = acc[r] * sSw[wave][el];
      }
    }
    __builtin_amdgcn_wave_barrier();

    // ---- balanced scatter: all 32 lanes, consecutive lanes -> consecutive
    // addresses (16-lane 64B groups) for atomic coalescing
    for (int t = lane; t < 16 * RHO_PER_ATOM; t += 32) {
      int el  = t / RHO_PER_ATOM;
      int rem = t - el * RHO_PER_ATOM;     // c*16 + mm
      int e   = base + el;
      if (e < nE) {
        int c = rem >> 4, mm = rem & 15;
        float v = sXs[wave][el][c] * sYe[wave][el][mm];
        atomicAdd(rho + (long)sSrc[wave][el] * RHO_PER_ATOM + rem, v);
      }
    }
    __builtin_amdgcn_wave_barrier();
  }
}

// ---------------- per-atom contractions -------------------------------------
__global__ void contract(const float* __restrict__ rho,
                         const float* __restrict__ species,
                         const float* __restrict__ Wg,
                         float* __restrict__ out, int nA) {
  __shared__ float sWg[520];
  __shared__ float sR[CBLK * 113];   // stride 113 (odd) -> conflict-free banks
  const int tid = threadIdx.x;
  for (int t = tid; t < 520; t += blockDim.x) sWg[t] = Wg[t];
  const int a = blockIdx.x * CBLK + tid;
  float* R = &sR[tid * 113];
  if (a < nA)
    for (int i = 0; i < RHO_PER_ATOM; ++i) R[i] = rho[(long)a * RHO_PER_ATOM + i];
  __syncthreads();
  if (a >= nA) return;

  float* o = out + (long)a * OUT_PER_ATOM;
  for (int c = 0; c < 16; ++c) o[c] = species[(long)a * AFVS + c];
  for (int c = 0; c < NCH; ++c) o[16 + c] = R[c * 16 + 0];

  // pair contractions (28 pairs)
  int p = 0;
  for (int i = 0; i < NCH; ++i)
    for (int j = i; j < NCH; ++j) {
      const float* Ri = R + i * 16;
      const float* Rj = R + j * 16;
      float s1 = 0.f, s2 = 0.f, s3 = 0.f;
#pragma unroll
      for (int t = 0; t < 3; ++t) s1 += Ri[1 + t] * Rj[1 + t];
#pragma unroll
      for (int t = 0; t < 5; ++t) s2 += Ri[4 + t] * Rj[4 + t];
#pragma unroll
      for (int t = 0; t < 7; ++t) s3 += Ri[9 + t] * Rj[9 + t];
      o[23 + p] = s1; o[51 + p] = s2; o[79 + p] = s3; ++p;
    }

  // triplet contractions (84 triplets)
  const float* W112 = sWg;         // [n<3][o<3][m<5]
  const float* W222 = sWg + 45;    // [n<5][o<5][m<5]
  const float* W123 = sWg + 170;   // [n<3][o<5][m<7]
  const float* W233 = sWg + 275;   // [n<5][o<7][m<7]
  int t3 = 0;
  for (int i = 0; i < NCH; ++i)
    for (int j = i; j < NCH; ++j)
      for (int k = j; k < NCH; ++k) {
        const float* Ri = R + i * 16;
        const float* Rj = R + j * 16;
        const float* Rk = R + k * 16;
        float a211 = 0.f, a222 = 0.f, a312 = 0.f, a323 = 0.f;
        // xi211 = sum r2[i][m] r1[j][n] r1[k][o] W112[n][o][m]
        for (int n = 0; n < 3; ++n)
          for (int o2 = 0; o2 < 3; ++o2) {
            float pr = Rj[1 + n] * Rk[1 + o2];
            const float* wr = W112 + n * 15 + o2 * 5;
#pragma unroll
            for (int mm = 0; mm < 5; ++mm) a211 += pr * Ri[4 + mm] * wr[mm];
          }
        // xi222 = sum r2[i][m] r2[j][n] r2[k][o] W222[n][o][m]
        for (int n = 0; n < 5; ++n)
          for (int o2 = 0; o2 < 5; ++o2) {
            float pr = Rj[4 + n] * Rk[4 + o2];
            const float* wr = W222 + n * 25 + o2 * 5;
#pragma unroll
            for (int mm = 0; mm < 5; ++mm) a222 += pr * Ri[4 + mm] * wr[mm];
          }
        // xi312 = sum r3[i][m] r1[j][n] r2[k][o] W123[n][o][m]
        for (int n = 0; n < 3; ++n)
          for (int o2 = 0; o2 < 5; ++o2) {
            float pr = Rj[1 + n] * Rk[4 + o2];
            const float* wr = W123 + n * 35 + o2 * 7;
#pragma unroll
            for (int mm = 0; mm < 7; ++mm) a312 += pr * Ri[9 + mm] * wr[mm];
          }
        // xi323 = sum r3[i][m] r2[j][n] r3[k][o] W233[n][o][m]
        for (int n = 0; n < 5; ++n)
          for (int o2 = 0; o2 < 7; ++o2) {
            float pr = Rj[4 + n] * Rk[9 + o2];
            const float* wr = W233 + n * 49 + o2 * 7;
#pragma unroll
            for (int mm = 0; mm < 7; ++mm) a323 += pr * Ri[9 + mm] * wr[mm];
          }
        o[107 + t3] = a211; o[191 + t3] = a222; o[275 + t3] = a312; o[359 + t3] = a323;
        ++t3;
      }
}

// ---------------- launch -----------------------------------------------------
extern "C" void kernel_launch(void* const* d_in, const int* in_sizes, int n_in,
                              void* d_out, int out_size, void* d_ws, size_t ws_size,
                              hipStream_t stream) {
  (void)n_in; (void)out_size; (void)ws_size;
  const float* species = (const float*)d_in[0];
  const float* radial  = (const float*)d_in[1];
  const float* swit    = (const float*)d_in[2];
  const float* vec     = (const float*)d_in[3];
  const float* chemrad = (const float*)d_in[4];
  const int*   esrc    = (const int*)d_in[5];
  const int*   edst    = (const int*)d_in[6];

  const int nA = in_sizes[0] / AFVS;   // 50000
  const int nE = in_sizes[2];          // 800000 (switch is 1 per edge)
  const int nTiles = (nE + 15) / 16;   // 50000 WMMA tiles

  float* rho = (float*)d_ws;                        // [nA][112]
  float* Wg  = rho + (long)nA * RHO_PER_ATOM;       // [520]

  const long nrho = (long)nA * RHO_PER_ATOM;
  int zb = (int)((nrho + 255) / 256); if (zb > 8192) zb = 8192;
  zero_f32<<<zb, 256, 0, stream>>>(rho, nrho);
  gaunt_init<<<1, 256, 0, stream>>>(Wg);

  int eblocks = (nTiles + 3) / 4; if (eblocks > EDGE_BLOCKS) eblocks = EDGE_BLOCKS;
  edge_wmma<<<EDGE_BLOCKS, 128, 0, stream>>>(species, radial, swit, vec, chemrad,
                                             esrc, edst, rho, nE, nTiles);
  contract<<<(nA + CBLK - 1) / CBLK, CBLK, 0, stream>>>(rho, species, Wg,
                                                        (float*)d_out, nA);
}